// PointnetSAModuleMSG_64673617543472
// MI455X (gfx1250) — compile-verified
//
#include <hip/hip_runtime.h>
#include <hip/hip_bf16.h>

typedef __attribute__((ext_vector_type(2))) float v2f;
typedef __attribute__((ext_vector_type(8))) float v8f;

#define BDIM 8
#define NPTS 16384
#define NPOINT 512
#define EPSV 1e-5f

// ------------------------------------------------------------------
// Ball grouping: one wave32 per (batch, query). Scans the 16384 points
// in index order, keeps the first `nsample` within radius (matching the
// reference's masked-argsort), pads with the first neighbor's diff.
// Output layout: x0[b][s*3+d][p]  (channel-major for the GEMMs).
// ------------------------------------------------------------------
__global__ void ball_group_kernel(const float* __restrict__ xyz,
                                  float* __restrict__ x0,
                                  float radius2, int nsample) {
  const int q = blockIdx.x;      // query point index (< NPOINT)
  const int b = blockIdx.y;      // batch
  const int lane = threadIdx.x;  // 0..31 (wave32)
  const int C0 = 3 * nsample;

  const float* base = xyz + (size_t)b * NPTS * 3;
  const float qx = base[q * 3 + 0];
  const float qy = base[q * 3 + 1];
  const float qz = base[q * 3 + 2];

  float* outp = x0 + (size_t)b * C0 * NPOINT;

  int total = 0;
  int found = 0;
  float fdx = 0.f, fdy = 0.f, fdz = 0.f;

  for (int beg = 0; beg < NPTS && total < nsample; beg += 32) {
    const int i = beg + lane;
    const float dx = base[i * 3 + 0] - qx;
    const float dy = base[i * 3 + 1] - qy;
    const float dz = base[i * 3 + 2] - qz;
    const float d2 = dx * dx + dy * dy + dz * dz;
    const bool pred = d2 < radius2;
    const unsigned mask = (unsigned)__ballot(pred);
    if (mask) {
      const int rank = __popc(mask & ((1u << lane) - 1u));
      const int slot = total + rank;
      if (pred && slot < nsample) {
        outp[(size_t)(slot * 3 + 0) * NPOINT + q] = dx;
        outp[(size_t)(slot * 3 + 1) * NPOINT + q] = dy;
        outp[(size_t)(slot * 3 + 2) * NPOINT + q] = dz;
      }
      if (!found) {
        const int fl = __ffs(mask) - 1;
        fdx = __shfl(dx, fl, 32);
        fdy = __shfl(dy, fl, 32);
        fdz = __shfl(dz, fl, 32);
        found = 1;
      }
      total += __popc(mask);
    }
  }
  // pad remaining slots with the first in-radius neighbor
  const int slot = total + lane;
  if (slot < nsample) {
    outp[(size_t)(slot * 3 + 0) * NPOINT + q] = fdx;
    outp[(size_t)(slot * 3 + 1) * NPOINT + q] = fdy;
    outp[(size_t)(slot * 3 + 2) * NPOINT + q] = fdz;
  }
}

// ------------------------------------------------------------------
// Fold inference batch-norm into per-channel scale/bias:
//   y = (Wx + b - m) * rsqrt(v+eps) * g + beta = (Wx)*sc + bi
// ------------------------------------------------------------------
__global__ void fold_bn_kernel(const float* __restrict__ b, const float* __restrict__ g,
                               const float* __restrict__ beta, const float* __restrict__ m,
                               const float* __restrict__ v,
                               float* __restrict__ sc, float* __restrict__ bi, int n) {
  const int i = blockIdx.x * blockDim.x + threadIdx.x;
  if (i < n) {
    const float s = g[i] * rsqrtf(v[i] + EPSV);
    sc[i] = s;
    bi[i] = (b[i] - m[i]) * s + beta[i];
  }
}

__global__ void copy_new_xyz_kernel(const float* __restrict__ xyz, float* __restrict__ out) {
  const int i = blockIdx.x * blockDim.x + threadIdx.x;
  if (i < BDIM * NPOINT * 3) {
    const int b = i / (NPOINT * 3);
    const int rem = i % (NPOINT * 3);
    out[i] = xyz[(size_t)b * NPTS * 3 + rem];
  }
}

// ------------------------------------------------------------------
// Generic batched f32 GEMM on V_WMMA_F32_16X16X4_F32.
// C[bz] = epilogue(alpha * A[bz] * op(B[bz]))
//   block tile 64(M) x 128(N), BK=16, 256 threads = 8 waves,
//   each wave: 16(M) x 64(N) via 4 accumulators of 16x16.
// Epilogue: optional per-row scale/bias (+ReLU), optional per-col bias.
// All problem dims here are exact multiples of the tile sizes.
// ------------------------------------------------------------------
#define GBM 64
#define GBN 128
#define GBK 16

__global__ __launch_bounds__(256) void gemm_wmma_f32(
    const float* __restrict__ A, const float* __restrict__ B, float* __restrict__ C,
    int M, int N, int K, int lda, int ldb, int ldc,
    long sA, long sB, long sC, int transB, float alpha,
    const float* __restrict__ rowScale, const float* __restrict__ rowBias,
    const float* __restrict__ colBias, int doRelu) {
  __shared__ float As[GBK][GBM + 1];   // As[k][m], +1 pad vs write conflicts
  __shared__ float Bs[GBK][GBN + 1];   // Bs[k][n], +1 pad vs transB write conflicts

  const int tid = threadIdx.x;
  const int lane = tid & 31;
  const int wid = tid >> 5;
  const int wm = wid & 3;   // wave row tile 0..3
  const int wn = wid >> 2;  // wave col tile 0..1
  const int r = lane & 15;
  const int half = lane >> 4;

  const int bz = blockIdx.z;
  const int row0 = blockIdx.y * GBM;
  const int col0 = blockIdx.x * GBN;

  const float* Ab = A + (size_t)bz * sA;
  const float* Bb = B + (size_t)bz * sB;
  float* Cb = C + (size_t)bz * sC;

  v8f acc[4] = {};

  for (int k0 = 0; k0 < K; k0 += GBK) {
    // stage A tile transposed: As[k][m] = A[(row0+m)*lda + k0+k]
#pragma unroll
    for (int i = 0; i < (GBM * GBK) / 256; ++i) {
      const int e = tid + i * 256;
      const int k = e & 15;
      const int m = e >> 4;
      As[k][m] = Ab[(size_t)(row0 + m) * lda + (k0 + k)];
    }
    if (transB) {
#pragma unroll
      for (int i = 0; i < (GBN * GBK) / 256; ++i) {
        const int e = tid + i * 256;
        const int k = e & 15;
        const int n = e >> 4;
        Bs[k][n] = Bb[(size_t)(col0 + n) * ldb + (k0 + k)];
      }
    } else {
#pragma unroll
      for (int i = 0; i < (GBN * GBK) / 256; ++i) {
        const int e = tid + i * 256;
        const int n = e & 127;
        const int k = e >> 7;
        Bs[k][n] = Bb[(size_t)(k0 + k) * ldb + (col0 + n)];
      }
    }
    __syncthreads();

#pragma unroll
    for (int kk = 0; kk < GBK; kk += 4) {
      // A fragment 16x4: lane r holds row M=r; v[j] = A[r][2*half+j]
      v2f a;
      a[0] = As[kk + 2 * half + 0][wm * 16 + r];
      a[1] = As[kk + 2 * half + 1][wm * 16 + r];
#pragma unroll
      for (int t = 0; t < 4; ++t) {
        // B fragment 4x16: lane r holds col N=r; v[j] = B[2*half+j][r]
        v2f bf;
        bf[0] = Bs[kk + 2 * half + 0][wn * 64 + t * 16 + r];
        bf[1] = Bs[kk + 2 * half + 1][wn * 64 + t * 16 + r];
        acc[t] = __builtin_amdgcn_wmma_f32_16x16x4_f32(
            false, a, false, bf, (short)0, acc[t], false, false);
      }
    }
    __syncthreads();
  }

  // epilogue + store (C/D layout: VGPR i -> row i + 8*half, col = r)
#pragma unroll
  for (int t = 0; t < 4; ++t) {
    const int col = col0 + wn * 64 + t * 16 + r;
#pragma unroll
    for (int i = 0; i < 8; ++i) {
      const int row = row0 + wm * 16 + i + 8 * half;
      float v = acc[t][i] * alpha;
      if (rowScale) v = v * rowScale[row] + rowBias[row];
      if (colBias) v += colBias[col];
      if (doRelu) v = fmaxf(v, 0.f);
      Cb[(size_t)row * ldc + col] = v;
    }
  }
}

// ------------------------------------------------------------------
// Row softmax over 512 columns; one wave per row, 8 rows per block.
// ------------------------------------------------------------------
__global__ __launch_bounds__(256) void softmax_rows_512(float* __restrict__ S, int nrows) {
  const int lane = threadIdx.x & 31;
  const int wid = threadIdx.x >> 5;
  const int row = blockIdx.x * 8 + wid;
  if (row >= nrows) return;
  float* p = S + (size_t)row * 512;
  float vals[16];
  float mx = -3.4e38f;
#pragma unroll
  for (int j = 0; j < 16; ++j) {
    vals[j] = p[j * 32 + lane];
    mx = fmaxf(mx, vals[j]);
  }
  for (int off = 16; off >= 1; off >>= 1) mx = fmaxf(mx, __shfl_xor(mx, off, 32));
  float sum = 0.f;
#pragma unroll
  for (int j = 0; j < 16; ++j) {
    vals[j] = __expf(vals[j] - mx);
    sum += vals[j];
  }
  for (int off = 16; off >= 1; off >>= 1) sum += __shfl_xor(sum, off, 32);
  const float inv = 1.f / sum;
#pragma unroll
  for (int j = 0; j < 16; ++j) p[j * 32 + lane] = vals[j] * inv;
}

// ------------------------------------------------------------------
// Orchestration. Input order (setup_inputs dict order):
//  0: xyz
//  per branch (24 entries): 3x conv {W,b,g,beta,m,v}, then Wq,bq,Wk,bk,Wv,bv
// Output: [new_xyz (8*512*3)] ++ [feats (8,1024,512)], branch0 -> ch 0..511.
// ------------------------------------------------------------------
extern "C" void kernel_launch(void* const* d_in, const int* in_sizes, int n_in,
                              void* d_out, int out_size, void* d_ws, size_t ws_size,
                              hipStream_t stream) {
  (void)in_sizes; (void)n_in; (void)out_size; (void)ws_size;
  const float* xyz = (const float*)d_in[0];
  float* out = (float*)d_out;
  float* ws = (float*)d_ws;

  // workspace layout (float elements)
  float* x0 = ws;               // up to 8*96*512   = 393216
  float* h1 = x0 + 393216;      // 8*64*512         = 262144
  float* h2 = h1 + 262144;      // 8*128*512        = 524288
  float* X  = h2 + 524288;      // 8*512*512        = 2097152
  float* Q  = X + 2097152;
  float* Km = Q + 2097152;
  float* V  = Km + 2097152;
  float* S  = V + 2097152;
  float* sc1 = S + 2097152; float* bi1 = sc1 + 64;
  float* sc2 = bi1 + 64;    float* bi2 = sc2 + 128;
  float* sc3 = bi2 + 128;   float* bi3 = sc3 + 512;

  const long bs = 512L * 512L;  // per-batch stride of 512x512 matrices

  copy_new_xyz_kernel<<<(BDIM * NPOINT * 3 + 255) / 256, 256, 0, stream>>>(xyz, out);

  const float radii[2] = {0.2f, 0.4f};
  const int nsamples[2] = {16, 32};

  for (int br = 0; br < 2; ++br) {
    const int pbase = 1 + br * 24;
    const int ns = nsamples[br];
    const int C0 = 3 * ns;
    const float r2 = radii[br] * radii[br];

    // 1) ball grouping -> x0 [B][C0][NPOINT]
    {
      dim3 g(NPOINT, BDIM);
      ball_group_kernel<<<g, 32, 0, stream>>>(xyz, x0, r2, ns);
    }

    // 2) fold BN for the three conv layers
    fold_bn_kernel<<<1, 64, 0, stream>>>(
        (const float*)d_in[pbase + 1], (const float*)d_in[pbase + 2],
        (const float*)d_in[pbase + 3], (const float*)d_in[pbase + 4],
        (const float*)d_in[pbase + 5], sc1, bi1, 64);
    fold_bn_kernel<<<1, 128, 0, stream>>>(
        (const float*)d_in[pbase + 7], (const float*)d_in[pbase + 8],
        (const float*)d_in[pbase + 9], (const float*)d_in[pbase + 10],
        (const float*)d_in[pbase + 11], sc2, bi2, 128);
    fold_bn_kernel<<<2, 256, 0, stream>>>(
        (const float*)d_in[pbase + 13], (const float*)d_in[pbase + 14],
        (const float*)d_in[pbase + 15], (const float*)d_in[pbase + 16],
        (const float*)d_in[pbase + 17], sc3, bi3, 512);

    const float* W0 = (const float*)d_in[pbase + 0];
    const float* W1 = (const float*)d_in[pbase + 6];
    const float* W2 = (const float*)d_in[pbase + 12];

    // 3) MLP: h = relu(sc*(W x) + bi)
    {
      dim3 g(4, 1, BDIM);  // N=512/128, M=64/64
      gemm_wmma_f32<<<g, 256, 0, stream>>>(W0, x0, h1, 64, 512, C0, C0, 512, 512,
                                           0L, (long)C0 * 512, 64L * 512, 0, 1.f,
                                           sc1, bi1, nullptr, 1);
    }
    {
      dim3 g(4, 2, BDIM);
      gemm_wmma_f32<<<g, 256, 0, stream>>>(W1, h1, h2, 128, 512, 64, 64, 512, 512,
                                           0L, 64L * 512, 128L * 512, 0, 1.f,
                                           sc2, bi2, nullptr, 1);
    }
    {
      dim3 g(4, 8, BDIM);
      gemm_wmma_f32<<<g, 256, 0, stream>>>(W2, h2, X, 512, 512, 128, 128, 512, 512,
                                           0L, 128L * 512, bs, 0, 1.f,
                                           sc3, bi3, nullptr, 1);
    }

    // 4) attention: Q/K/V = X * W^T + b(col)   (einsum 'bcn,mn->bcm')
    const float* Wq = (const float*)d_in[pbase + 18];
    const float* bq = (const float*)d_in[pbase + 19];
    const float* Wk = (const float*)d_in[pbase + 20];
    const float* bk = (const float*)d_in[pbase + 21];
    const float* Wv = (const float*)d_in[pbase + 22];
    const float* bv = (const float*)d_in[pbase + 23];

    dim3 ga(4, 8, BDIM);
    gemm_wmma_f32<<<ga, 256, 0, stream>>>(X, Wq, Q, 512, 512, 512, 512, 512, 512,
                                          bs, 0L, bs, 1, 1.f, nullptr, nullptr, bq, 0);
    gemm_wmma_f32<<<ga, 256, 0, stream>>>(X, Wk, Km, 512, 512, 512, 512, 512, 512,
                                          bs, 0L, bs, 1, 1.f, nullptr, nullptr, bk, 0);
    gemm_wmma_f32<<<ga, 256, 0, stream>>>(X, Wv, V, 512, 512, 512, 512, 512, 512,
                                          bs, 0L, bs, 1, 1.f, nullptr, nullptr, bv, 0);

    // 5) scores: S = (1/sqrt(512)) * Q K^T  (transB, batched B)
    gemm_wmma_f32<<<ga, 256, 0, stream>>>(Q, Km, S, 512, 512, 512, 512, 512, 512,
                                          bs, bs, bs, 1, 0.04419417382f,
                                          nullptr, nullptr, nullptr, 0);

    // 6) softmax over last axis
    softmax_rows_512<<<(BDIM * 512) / 8, 256, 0, stream>>>(S, BDIM * 512);

    // 7) O = softmax(S) * V, written straight into the concatenated output
    float* Cout = out + 12288 + (size_t)br * 512 * 512;
    gemm_wmma_f32<<<ga, 256, 0, stream>>>(S, V, Cout, 512, 512, 512, 512, 512, 512,
                                          bs, bs, 1024L * 512, 0, 1.f,
                                          nullptr, nullptr, nullptr, 0);
  }
}